// SGL_88313117540474
// MI455X (gfx1250) — compile-verified
//
#include <hip/hip_runtime.h>
#include <hip/hip_bf16.h>

#define N_USERS   100000
#define N_ITEMS   50000
#define N_NODES   (N_USERS + N_ITEMS)       // 150000
#define EMBED_DIM 128
#define N_EDGES   3000000
#define N_LAYERS  3

#define TOTAL4    (N_NODES * (EMBED_DIM / 4))   // 4,800,000 float4
#define USER4     (N_USERS * (EMBED_DIM / 4))   // 3,200,000 float4
#define DEG4      (N_NODES / 4)                 // 37,500 int4 (N_NODES % 4 == 0)

#define THREADS   256
#define CHUNK     1024    // edges staged to LDS per round (each thread: one b128 per array)

// ---- gfx1250 async global->LDS path (guarded so compile never breaks) ----
#if defined(__has_builtin)
#  if __has_builtin(__builtin_amdgcn_global_load_async_to_lds_b128)
#    define HAVE_ASYNC_LDS 1
#  endif
#  if __has_builtin(__builtin_amdgcn_s_wait_asynccnt)
#    define HAVE_WAIT_ASYNC_BUILTIN 1
#  endif
#endif

// Matches the parameter type hipcc reported for the async-LDS builtin.
typedef int v4i __attribute__((vector_size(16)));

__device__ __forceinline__ void wait_asynccnt0() {
#if defined(HAVE_WAIT_ASYNC_BUILTIN)
  __builtin_amdgcn_s_wait_asynccnt(0);
#else
  asm volatile("s_wait_asynccnt 0" ::: "memory");
#endif
}

// ---------------- scatter-add: agg[dst] += acc[src], wave-per-edge ----------------
// First in the file so the disasm snippet shows this kernel.

__global__ void __launch_bounds__(THREADS)
k_scatter(const float* __restrict__ acc, float* __restrict__ agg,
          const int* __restrict__ src, const int* __restrict__ dst) {
  __shared__ int lds_src[2][CHUNK];
  __shared__ int lds_dst[2][CHUNK];
  const int tid  = threadIdx.x;
  const int lane = tid & 31;
  const int wave = tid >> 5;
  const int nchunks = (N_EDGES + CHUNK - 1) / CHUNK;
  const int off  = tid * 4;   // dword offset this thread stages (16B aligned)

  // Stage (src,dst) indices for chunk `ch` into LDS buffer `b` (async DMA).
  auto stage = [&](int ch, int b) {
    const int cbase = ch * CHUNK;
    const int ccnt  = (N_EDGES - cbase < CHUNK) ? (N_EDGES - cbase) : CHUNK;
    if (off < ccnt) {
#if defined(HAVE_ASYNC_LDS)
      __builtin_amdgcn_global_load_async_to_lds_b128(
          (v4i*)(src + cbase + off), (v4i*)(&lds_src[b][off]), 0, 0);
      __builtin_amdgcn_global_load_async_to_lds_b128(
          (v4i*)(dst + cbase + off), (v4i*)(&lds_dst[b][off]), 0, 0);
#else
      *(int4*)(&lds_src[b][off]) = *(const int4*)(src + cbase + off);
      *(int4*)(&lds_dst[b][off]) = *(const int4*)(dst + cbase + off);
#endif
    }
  };

  int c = blockIdx.x;
  if (c >= nchunks) return;

  stage(c, 0);                      // prologue: fill buffer 0
  int buf = 0;

  for (; c < nchunks; c += gridDim.x) {
    wait_asynccnt0();               // own async stages of `buf` complete
    __syncthreads();                // all waves' stages visible; prev readers done

    const int next = c + (int)gridDim.x;
    if (next < nchunks) stage(next, buf ^ 1);   // overlap DMA with compute

    const int base = c * CHUNK;
    const int cnt  = (N_EDGES - base < CHUNK) ? (N_EDGES - base) : CHUNK;

    // Each wave takes every 8th edge; lane covers 4 columns (32*4 = 128).
#pragma unroll 2
    for (int e = wave; e < cnt; e += THREADS / 32) {
      const int s = lds_src[buf][e];     // lane-uniform LDS broadcast
      const int d = lds_dst[buf][e];
      const float4 v = *(const float4*)(acc + (size_t)s * EMBED_DIM + lane * 4);
      float* o = agg + (size_t)d * EMBED_DIM + lane * 4;
      unsafeAtomicAdd(o + 0, v.x);       // global_atomic_add_f32, non-returning
      unsafeAtomicAdd(o + 1, v.y);
      unsafeAtomicAdd(o + 2, v.z);
      unsafeAtomicAdd(o + 3, v.w);
    }
    __syncthreads();                // all reads of `buf` done before re-staging it
    buf ^= 1;
  }
}

// ---------------- elementwise kernels ----------------

__global__ void k_zero4(float4* __restrict__ p, int n4) {
  int i = blockIdx.x * blockDim.x + threadIdx.x;
  if (i < n4) p[i] = make_float4(0.f, 0.f, 0.f, 0.f);
}

__global__ void k_degree(const int* __restrict__ dst, int* __restrict__ deg) {
  int e = blockIdx.x * blockDim.x + threadIdx.x;
  if (e < N_EDGES) atomicAdd(&deg[dst[e]], 1);
}

__global__ void k_invdeg(const int* __restrict__ deg, float* __restrict__ inv) {
  int i = blockIdx.x * blockDim.x + threadIdx.x;
  if (i < N_NODES) {
    int d = deg[i];
    inv[i] = (d > 0) ? (1.0f / (float)d) : 0.0f;
  }
}

// acc0 = x = cat(user,item); out = 0.25*x  (1/(L+1) folded in)
__global__ void k_init(const float4* __restrict__ u, const float4* __restrict__ it,
                       float4* __restrict__ acc, float4* __restrict__ out) {
  int i = blockIdx.x * blockDim.x + threadIdx.x;
  if (i >= TOTAL4) return;
  float4 v = (i < USER4) ? u[i] : it[i - USER4];
  acc[i] = v;
  out[i] = make_float4(0.25f * v.x, 0.25f * v.y, 0.25f * v.z, 0.25f * v.w);
}

// buf = buf * inv_deg(row) (in place -> next layer's acc); out += 0.25*buf
__global__ void k_scale_accum(float4* __restrict__ buf, const float* __restrict__ inv,
                              float4* __restrict__ out) {
  int i = blockIdx.x * blockDim.x + threadIdx.x;
  if (i >= TOTAL4) return;
  float w = inv[i >> 5];   // 32 float4 per 128-wide row
  float4 a = buf[i];
  a.x *= w; a.y *= w; a.z *= w; a.w *= w;
  buf[i] = a;
  float4 o = out[i];
  o.x += 0.25f * a.x; o.y += 0.25f * a.y; o.z += 0.25f * a.z; o.w += 0.25f * a.w;
  out[i] = o;
}

// ---------------- launch ----------------

extern "C" void kernel_launch(void* const* d_in, const int* in_sizes, int n_in,
                              void* d_out, int out_size, void* d_ws, size_t ws_size,
                              hipStream_t stream) {
  const float4* user = (const float4*)d_in[0];
  const float4* item = (const float4*)d_in[1];
  const int*    edges = (const int*)d_in[2];
  const int*    src = edges;             // edge_index[0]
  const int*    dst = edges + N_EDGES;   // edge_index[1]
  float4*       out4 = (float4*)d_out;

  char* ws = (char*)d_ws;
  int*   deg  = (int*)ws;                                        // N_NODES ints
  float* inv  = (float*)(ws + (size_t)N_NODES * 4);              // N_NODES floats
  float* bufA = (float*)(ws + (size_t)N_NODES * 8);              // N_NODES*128 floats
  float* bufB = bufA + (size_t)N_NODES * EMBED_DIM;              // N_NODES*128 floats

  const int gN4   = (TOTAL4 + THREADS - 1) / THREADS;   // 18750
  const int gDeg4 = (DEG4 + THREADS - 1) / THREADS;
  const int gE    = (N_EDGES + THREADS - 1) / THREADS;
  const int gN    = (N_NODES + THREADS - 1) / THREADS;

  const int nchunks  = (N_EDGES + CHUNK - 1) / CHUNK;   // 2930
  const int gScatter = (nchunks + 1) / 2;               // ~2 chunks/block -> pipeline engages

  // degree -> inv_deg
  k_zero4<<<gDeg4, THREADS, 0, stream>>>((float4*)deg, DEG4);
  k_degree<<<gE, THREADS, 0, stream>>>(dst, deg);
  k_invdeg<<<gN, THREADS, 0, stream>>>(deg, inv);

  // acc0 = x, out = x/4
  k_init<<<gN4, THREADS, 0, stream>>>(user, item, (float4*)bufA, out4);

  float* cur = bufA;
  float* nxt = bufB;
  for (int l = 0; l < N_LAYERS; ++l) {
    k_zero4<<<gN4, THREADS, 0, stream>>>((float4*)nxt, TOTAL4);
    k_scatter<<<gScatter, THREADS, 0, stream>>>(cur, nxt, src, dst);
    k_scale_accum<<<gN4, THREADS, 0, stream>>>((float4*)nxt, inv, out4);
    float* t = cur; cur = nxt; nxt = t;
  }
}